// TpsWarp_44220983280422
// MI455X (gfx1250) — compile-verified
//
#include <hip/hip_runtime.h>

// CDNA5 / gfx1250 TPS warp.
//  kernel 1: build augmented TPS system  [K P ; P^T 0 | rhs]   (B x 259 x 261)
//  kernel 2: Gauss-Jordan solve with partial pivoting, 1 workgroup per batch
//  kernel 3: grid evaluation via V_WMMA_F32_16X16X4_F32 (f32 WMMA), M=16 grid
//            points x K=4 control points per step, N=16 (cols 0,1 = x,y out)

typedef __attribute__((ext_vector_type(2))) float v2f;
typedef __attribute__((ext_vector_type(8))) float v8f;

#define N_CTRL 256
#define NP     259          // n + 3
#define NAUG   261          // NP + 2 rhs columns
#define LDA    264          // padded row stride in floats
#define S_GRID 256
#define S2     (S_GRID * S_GRID)

// Hardware V_SQRT_F32 (~1 ulp). Inputs here are in [0, 8]; no denormal or
// overflow concerns, so skip the correctly-rounded libm expansion.
__device__ __forceinline__ float fast_sqrtf(float x) {
  return __builtin_amdgcn_sqrtf(x);
}

// ---------------------------------------------------------------------------
// Kernel 1: build the augmented system for all batches.
// ---------------------------------------------------------------------------
__global__ void tps_build(const float* __restrict__ src,
                          const float* __restrict__ dst,
                          float* __restrict__ aug, int B) {
  int idx = blockIdx.x * blockDim.x + threadIdx.x;
  int total = B * NP * NAUG;
  if (idx >= total) return;
  int j = idx % NAUG;
  int t = idx / NAUG;
  int i = t % NP;
  int b = t / NP;
  const float* sb = src + (size_t)b * N_CTRL * 2;
  const float* db = dst + (size_t)b * N_CTRL * 2;
  float v;
  if (j < N_CTRL) {
    if (i < N_CTRL) {                       // K block: |src_i - src_j|
      float dx = sb[i * 2 + 0] - sb[j * 2 + 0];
      float dy = sb[i * 2 + 1] - sb[j * 2 + 1];
      v = fast_sqrtf(dx * dx + dy * dy);    // sqrt(0)=0 matches safe_norm
    } else {                                // P^T block rows: 1, x, y
      int r = i - N_CTRL;
      v = (r == 0) ? 1.0f : sb[j * 2 + (r - 1)];
    }
  } else if (j < NP) {                      // P block cols: 1, x, y  (or 0)
    int c = j - N_CTRL;
    v = (i < N_CTRL) ? ((c == 0) ? 1.0f : sb[i * 2 + (c - 1)]) : 0.0f;
  } else {                                  // rhs columns
    int c = j - NP;
    v = (i < N_CTRL) ? db[i * 2 + c] : 0.0f;
  }
  aug[(size_t)b * NP * LDA + (size_t)i * LDA + j] = v;
}

// ---------------------------------------------------------------------------
// Kernel 2: Gauss-Jordan with partial pivoting. One 256-thread block / batch.
// ---------------------------------------------------------------------------
__global__ void __launch_bounds__(256)
tps_solve(float* __restrict__ aug, float* __restrict__ wv) {
  __shared__ float smax[256];
  __shared__ int   sidx[256];
  __shared__ float rowk[NAUG];
  __shared__ int   spiv;
  const int tid = threadIdx.x;
  float* A = aug + (size_t)blockIdx.x * NP * LDA;

  for (int k = 0; k < NP; ++k) {
    // --- partial pivot: argmax |A[i][k]|, i in [k, NP) ---
    float best = -1.0f; int bi = k;
    for (int i = k + tid; i < NP; i += 256) {
      float v = fabsf(A[(size_t)i * LDA + k]);
      if (v > best) { best = v; bi = i; }
    }
    smax[tid] = best; sidx[tid] = bi;
    __syncthreads();
    for (int off = 128; off > 0; off >>= 1) {
      if (tid < off && smax[tid + off] > smax[tid]) {
        smax[tid] = smax[tid + off]; sidx[tid] = sidx[tid + off];
      }
      __syncthreads();
    }
    if (tid == 0) spiv = sidx[0];
    __syncthreads();
    const int piv = spiv;

    // --- swap rows k <-> piv (cols < k are already zero in both) ---
    if (piv != k) {
      for (int j = k + tid; j < NAUG; j += 256) {
        float a = A[(size_t)k   * LDA + j];
        float p = A[(size_t)piv * LDA + j];
        A[(size_t)k   * LDA + j] = p;
        A[(size_t)piv * LDA + j] = a;
      }
    }
    __syncthreads();

    // --- normalize pivot row, stage it in LDS ---
    const float inv = 1.0f / A[(size_t)k * LDA + k];
    for (int j = k + tid; j < NAUG; j += 256) {
      float r = A[(size_t)k * LDA + j] * inv;
      A[(size_t)k * LDA + j] = r;
      rowk[j] = r;
    }
    __syncthreads();

    // --- eliminate column k from all other rows ---
    for (int i = tid; i < NP; i += 256) {
      if (i == k) continue;
      float f = A[(size_t)i * LDA + k];
      if (f != 0.0f) {
        for (int j = k; j < NAUG; ++j)
          A[(size_t)i * LDA + j] -= f * rowk[j];
      }
    }
    __syncthreads();
  }

  // --- solution = rhs columns of the reduced system ---
  for (int k = tid; k < NP; k += 256) {
    wv[((size_t)blockIdx.x * NP + k) * 2 + 0] = A[(size_t)k * LDA + NP + 0];
    wv[((size_t)blockIdx.x * NP + k) * 2 + 1] = A[(size_t)k * LDA + NP + 1];
  }
}

// ---------------------------------------------------------------------------
// Kernel 3: evaluate warp on the 256x256 grid with f32 WMMA.
//   wave32: A = 16x4 tile (16 grid points x 4 control points), lane l holds
//   A[M=l&15][K = vgpr + (l>>4)*2]; B = 4x16 with N = l&15 (only N<2 nonzero);
//   C/D 16x16: lane N holds column N for rows r + (l>=16)*8.
// ---------------------------------------------------------------------------
__global__ void __launch_bounds__(256)
tps_eval(const float* __restrict__ src,
         const float* __restrict__ wv,
         float* __restrict__ out) {
  __shared__ float2 sp[N_CTRL];      // control points of this batch
  __shared__ float  wvl[260 * 2];    // weights, row 259 zero-padded

  const int b   = blockIdx.y;
  const int tid = threadIdx.x;

  {
    const float* sb = src + (size_t)b * N_CTRL * 2;
    float2 p; p.x = sb[tid * 2 + 0]; p.y = sb[tid * 2 + 1];
    sp[tid] = p;
    const float* wb = wv + (size_t)b * NP * 2;
    for (int i = tid; i < 520; i += 256)
      wvl[i] = (i < NP * 2) ? wb[i] : 0.0f;
  }
  __syncthreads();

  const int lane = tid & 31;
  const int wave = tid >> 5;
  const int m    = lane & 15;        // A row / C column index
  const int half = lane >> 4;        // K-pair selector
  const int gp   = (blockIdx.x * 8 + wave) * 16 + m;   // grid point of A rows

  const float step = 2.0f / (float)(S_GRID - 1);
  const float gx = (float)(gp % S_GRID) * step - 1.0f;
  const float gy = (float)(gp / S_GRID) * step - 1.0f;

  const float colmask = (m < 2) ? 1.0f : 0.0f;  // only N=0,1 of B are nonzero
  const int   ncl     = m & 1;                  // which wv component (x or y)

  v8f acc = {};
  // main loop: 4 control points per WMMA, each lane computes 2 radial values
#pragma unroll 4
  for (int k0 = 0; k0 < N_CTRL; k0 += 4) {
    const int c0 = k0 + half * 2;
    float2 p0 = sp[c0];
    float2 p1 = sp[c0 + 1];
    float dx0 = gx - p0.x, dy0 = gy - p0.y;
    float dx1 = gx - p1.x, dy1 = gy - p1.y;
    v2f a;
    a.x = fast_sqrtf(dx0 * dx0 + dy0 * dy0);
    a.y = fast_sqrtf(dx1 * dx1 + dy1 * dy1);
    v2f bb;
    bb.x = wvl[c0 * 2 + ncl] * colmask;
    bb.y = wvl[(c0 + 1) * 2 + ncl] * colmask;
    acc = __builtin_amdgcn_wmma_f32_16x16x4_f32(
        /*neg_a=*/false, a, /*neg_b=*/false, bb,
        /*c_mod=*/(short)0, acc, /*reuse_a=*/false, /*reuse_b=*/false);
  }
  // affine tail: virtual K rows 256..259 carry [1, gx, gy, 0]
  {
    const int c0 = N_CTRL + half * 2;
    v2f a;
    a.x = (half == 0) ? 1.0f : gy;
    a.y = (half == 0) ? gx   : 0.0f;
    v2f bb;
    bb.x = wvl[c0 * 2 + ncl] * colmask;       // wv rows 256 / 258
    bb.y = wvl[(c0 + 1) * 2 + ncl] * colmask; // wv rows 257 / 259 (zero)
    acc = __builtin_amdgcn_wmma_f32_16x16x4_f32(
        false, a, false, bb, (short)0, acc, false, false);
  }

  // lanes with column index 0/1 hold the x/y outputs for 8 grid rows each
  if (m < 2) {
    const int   c   = m;                                  // 0 = x, 1 = y
    const int   gpb = (blockIdx.x * 8 + wave) * 16 + half * 8;
    float* ob = out + ((size_t)(b * 2 + c)) * S2 + gpb;
    ob[0] = acc[0]; ob[1] = acc[1]; ob[2] = acc[2]; ob[3] = acc[3];
    ob[4] = acc[4]; ob[5] = acc[5]; ob[6] = acc[6]; ob[7] = acc[7];
  }
}

// ---------------------------------------------------------------------------
extern "C" void kernel_launch(void* const* d_in, const int* in_sizes, int n_in,
                              void* d_out, int out_size, void* d_ws, size_t ws_size,
                              hipStream_t stream) {
  const float* src = (const float*)d_in[0];
  const float* dst = (const float*)d_in[1];
  // d_in[2] is the scalar s (== 256), baked in as S_GRID at compile time.
  const int B = in_sizes[0] / (N_CTRL * 2);

  float* aug = (float*)d_ws;                         // B * NP * LDA floats
  float* wvp = aug + (size_t)B * NP * LDA;           // B * NP * 2 floats

  {
    int total = B * NP * NAUG;
    tps_build<<<dim3((total + 255) / 256), dim3(256), 0, stream>>>(src, dst, aug, B);
  }
  tps_solve<<<dim3(B), dim3(256), 0, stream>>>(aug, wvp);
  tps_eval<<<dim3(S2 / (8 * 16), B), dim3(256), 0, stream>>>(src, wvp, (float*)d_out);
}